// UltraTinyGRU_2216203125552
// MI455X (gfx1250) — compile-verified
//
#include <hip/hip_runtime.h>

// CDNA5 / gfx1250 fused UltraTinyGRU:
//   - single pass over x (134 MB -> ~5.8us HBM floor @ 23.3 TB/s)
//   - input proj  : 4x chained V_WMMA_F32_16X16X4_F32 (K=16)
//   - hidden proj : 1x  V_WMMA_F32_16X16X4_F32 (K=4)
//   - branch-free gate math (v_exp_f32/v_rcp_f32) in WMMA D layout
//   - cross-column moves via ds_bpermute with hoisted addresses
//   - h D-layout -> A-layout transpose staged through 1KB LDS
// One wave (32 lanes) owns 16 batch chains; 256 single-wave workgroups.
// t-loop unrolled x2 with two statically-named A-operand register sets
// (no runtime-indexed arrays -> no v_movrel, loads stay pipelined).

typedef float v2f __attribute__((ext_vector_type(2)));
typedef float v8f __attribute__((ext_vector_type(8)));

#define T_STEPS 512
#define NF 16      // input features (= WMMA K total, 4 chunks of 4)
#define NH 4       // hidden
#define NG 12      // 3*NH gate columns (cols 12..15 zero-padded)

#define LOG2E 1.442695041f

__device__ __forceinline__ float fast_sigmoid(float x) {
  // 1/(1+2^(-x*log2e)) : v_mul + v_exp + v_add + v_rcp, no branches
  const float e = __builtin_amdgcn_exp2f(x * -LOG2E);
  return __builtin_amdgcn_rcpf(1.0f + e);
}

__device__ __forceinline__ float fast_tanh(float x) {
  // 1 - 2/(2^(2x*log2e)+1) : v_mul + v_exp + v_add + v_rcp + v_fma
  const float e = __builtin_amdgcn_exp2f(x * (2.0f * LOG2E));
  return __builtin_fmaf(-2.0f, __builtin_amdgcn_rcpf(e + 1.0f), 1.0f);
}

__global__ __launch_bounds__(32) void gru_fused_kernel(
    const float* __restrict__ x,      // [B, T, 16]
    const float* __restrict__ w_ih,   // [12, 16]
    const float* __restrict__ w_hh,   // [12, 4]
    const float* __restrict__ b_ih,   // [12]
    const float* __restrict__ b_hh,   // [12]
    const float* __restrict__ fc_w,   // [1, 4]
    const float* __restrict__ fc_b,   // [1]
    float* __restrict__ out)          // [B]
{
  __shared__ float hbuf[16 * 16];     // h staging: [chain][col], 1 KB

  const int lane = threadIdx.x;       // 0..31 (wave32)
  const int m    = lane & 15;         // tile row (A) / column index (B,C,D)
  const int half = lane >> 4;         // 0 or 1
  const int b0   = blockIdx.x * 16;   // first batch chain of this wave

  // loop-invariant ds_bpermute byte addresses (dst[lane] = src[addr>>2])
  const int pa_x8 = (lane ^ 8) << 2;
  const int pa_x4 = (lane ^ 4) << 2;

  // ---------------- loop-invariant WMMA operands ----------------
  // B layout (4x16 f32, 2 VGPRs): vgpr j -> row K=j+2*half at this lane half
  const int   n     = m;                              // gate column for B/C/D
  const int   nc    = (n < NG) ? n : 0;               // clamp: no OOB reads
  const float nmask = (n < NG) ? 1.0f : 0.0f;         // zero-pad cols 12..15

  v2f Bih[4];                                         // w_ih^T, 4 K-chunks
  #pragma unroll
  for (int c = 0; c < 4; ++c) {
    const int k0 = 2 * half;
    Bih[c].x = nmask * w_ih[nc * NF + (4 * c + k0 + 0)];
    Bih[c].y = nmask * w_ih[nc * NF + (4 * c + k0 + 1)];
  }
  v2f Bhh;                                            // w_hh^T (K=4 exactly)
  Bhh.x = nmask * w_hh[nc * NH + (2 * half + 0)];
  Bhh.y = nmask * w_hh[nc * NH + (2 * half + 1)];

  const float cb_ih = nmask * b_ih[nc];               // biases ride the C operand
  const float cb_hh = nmask * b_hh[nc];
  v8f Cih, Chh;
  #pragma unroll
  for (int j = 0; j < 8; ++j) { Cih[j] = cb_ih; Chh[j] = cb_hh; }

  // ---------------- per-lane x stream ----------------
  // A layout (16x4 f32): lane m holds K = 2*half, 2*half+1 -> one float2 per chunk
  const float* xb = x + ((size_t)(b0 + m) * T_STEPS) * NF + 2 * half;

  v2f ah; ah.x = 0.0f; ah.y = 0.0f;                   // h0 = 0 in A layout
  float hprev[8];
  #pragma unroll
  for (int j = 0; j < 8; ++j) hprev[j] = 0.0f;        // h0 = 0 in D layout

  const int stbase = half * 128 + n;                  // LDS store: chain*16 + col
  const int ldoff  = m * 16 + 2 * half;               // LDS load : h[m][2*half..+1]

  // two statically-named double-buffer register sets (no runtime indexing)
  v2f a0_0, a0_1, a0_2, a0_3;                         // even t
  v2f a1_0, a1_1, a1_2, a1_3;                         // odd  t

  {                                                   // preload t = 0
    const float* p = xb;
    a0_0 = *(const v2f*)(p + 0);  a0_1 = *(const v2f*)(p + 4);
    a0_2 = *(const v2f*)(p + 8);  a0_3 = *(const v2f*)(p + 12);
  }

  // one GRU step: WMMAs + gate math on a given A-operand set
  auto step = [&](v2f& c0, v2f& c1, v2f& c2, v2f& c3) {
    v8f gi = Cih;                                     // gi = x_t*w_ih^T + b_ih
    gi = __builtin_amdgcn_wmma_f32_16x16x4_f32(false, c0, false, Bih[0], (short)0, gi, false, false);
    gi = __builtin_amdgcn_wmma_f32_16x16x4_f32(false, c1, false, Bih[1], (short)0, gi, false, false);
    gi = __builtin_amdgcn_wmma_f32_16x16x4_f32(false, c2, false, Bih[2], (short)0, gi, false, false);
    gi = __builtin_amdgcn_wmma_f32_16x16x4_f32(false, c3, false, Bih[3], (short)0, gi, false, false);
    // gh = h_{t-1}*w_hh^T + b_hh
    v8f gh = __builtin_amdgcn_wmma_f32_16x16x4_f32(false, ah, false, Bhh, (short)0, Chh, false, false);

    // D layout: VGPR j = chains {j, j+8}; lane column = gate (r:0-3 z:4-7 n:8-11)
    #pragma unroll
    for (int j = 0; j < 8; ++j) {
      const float g  = gi[j] + gh[j];
      const float s  = fast_sigmoid(g);               // valid at r,z columns
      const float r8 = __int_as_float(__builtin_amdgcn_ds_bpermute(pa_x8, __float_as_int(s)));
      const float np = __builtin_fmaf(r8, gh[j], gi[j]);   // n preact @ cols 8-11
      const float nn = fast_tanh(np);
      const float z4 = __int_as_float(__builtin_amdgcn_ds_bpermute(pa_x4, __float_as_int(s)));
      const float n8 = __int_as_float(__builtin_amdgcn_ds_bpermute(pa_x8, __float_as_int(nn)));
      const float hn = __builtin_fmaf(z4, hprev[j] - n8, n8);  // (1-z)*n + z*h
      hprev[j] = hn;
      hbuf[stbase + j * 16] = hn;                     // stage for A-layout gather
    }
    ah = *(const v2f*)&hbuf[ldoff];                   // D->A transpose (same-wave LDS)
  };

  for (int t = 0; t < T_STEPS; t += 2) {
    {                                                 // load t+1 (T even: always valid)
      const float* p = xb + (size_t)(t + 1) * NF;
      a1_0 = *(const v2f*)(p + 0);  a1_1 = *(const v2f*)(p + 4);
      a1_2 = *(const v2f*)(p + 8);  a1_3 = *(const v2f*)(p + 12);
    }
    __builtin_prefetch(xb + (size_t)(t + 16) * NF, 0, 3);  // stream ~1KB ahead, keep near
    step(a0_0, a0_1, a0_2, a0_3);                     // t

    if (t + 2 < T_STEPS) {                            // uniform branch
      const float* p = xb + (size_t)(t + 2) * NF;
      a0_0 = *(const v2f*)(p + 0);  a0_1 = *(const v2f*)(p + 4);
      a0_2 = *(const v2f*)(p + 8);  a0_3 = *(const v2f*)(p + 12);
    }
    step(a1_0, a1_1, a1_2, a1_3);                     // t+1
  }

  // final linear head: out[b] = h_T . fc_w + fc_b
  if (half == 0) {
    const float4 hv = *(const float4*)&hbuf[m * 16];
    out[b0 + m] = fc_b[0] + hv.x * fc_w[0] + hv.y * fc_w[1]
                          + hv.z * fc_w[2] + hv.w * fc_w[3];
  }
}

extern "C" void kernel_launch(void* const* d_in, const int* in_sizes, int n_in,
                              void* d_out, int out_size, void* d_ws, size_t ws_size,
                              hipStream_t stream) {
  const float* x    = (const float*)d_in[0];
  const float* w_ih = (const float*)d_in[1];
  const float* w_hh = (const float*)d_in[2];
  const float* b_ih = (const float*)d_in[3];
  const float* b_hh = (const float*)d_in[4];
  const float* fc_w = (const float*)d_in[5];
  const float* fc_b = (const float*)d_in[6];
  float* out = (float*)d_out;

  const int B = in_sizes[0] / (T_STEPS * NF);         // 4096
  const int nblocks = B / 16;                         // 16 chains per wave
  gru_fused_kernel<<<dim3(nblocks), dim3(32), 0, stream>>>(
      x, w_ih, w_hh, b_ih, b_hh, fc_w, fc_b, out);
}